// GCN_32349693673743
// MI455X (gfx1250) — compile-verified
//
#include <hip/hip_runtime.h>
#include <stdint.h>

// GCN with GraphConv(norm='none', weight=False, bias=False):
//   h_{l+1} = scatter_add_{dst}( gather_{src}( relu?(h_l) ) )
// Pure gather/scatter-add, D=48 f32. Feature table (19.2MB) is L2-resident on
// MI455X (192MB L2) -> latency-bound, so we hide gather latency with the CDNA5
// async global->LDS engine (ASYNCcnt) and double-buffered 32-edge tiles.

#define D_FEAT   48
#define ROW_BYTES 192          // 48 * 4B
#define LDS_ROW  208           // padded to 13 dword4 to de-stagger 64 LDS banks
#define TILE     32            // edges per wave-tile (1 edge per lane, wave32)
#define WPB      4             // waves per block
#define BLOCK    (WPB * 32)

typedef __attribute__((ext_vector_type(4))) float f4;
#define AS3 __attribute__((address_space(3)))

// ---------------------------------------------------------------- zero fill
__global__ __launch_bounds__(256) void gcn_zero(f4* __restrict__ p, int n4) {
  int i = blockIdx.x * 256 + threadIdx.x;
  const int stride = gridDim.x * 256;
  f4 z = 0.0f;
  for (; i < n4; i += stride) p[i] = z;
}

// ------------------------------------------------------- edge gather/scatter
// One edge per lane. Each wave owns a private double buffer in LDS:
//   stage(tile)  : 12x global_load_async_to_lds_b128 (192B row per lane)
//   consume(tile): 12x ds_load_b128 + 48x global_atomic_add_f32 per lane
template <int RELU>
__global__ __launch_bounds__(BLOCK) void gcn_edge_scatter(
    const float* __restrict__ h_in,
    const int*   __restrict__ src,
    const int*   __restrict__ dst,
    float*       __restrict__ h_out,
    int n_edges) {
  __shared__ __align__(16) char lds_raw[WPB * 2 * TILE * LDS_ROW];  // 53,248 B

  const int lane    = threadIdx.x & 31;
  const int wib     = threadIdx.x >> 5;
  const int wave    = blockIdx.x * WPB + wib;
  const int n_waves = gridDim.x * WPB;
  const int n_tiles = (n_edges + TILE - 1) / TILE;

  // Per-lane row bases inside this wave's two LDS buffers.
  AS3 char* lb   = (AS3 char*)lds_raw;
  AS3 char* row0 = lb + (uint32_t)(wib * 2 * TILE) * LDS_ROW + (uint32_t)lane * LDS_ROW;
  AS3 char* row1 = row0 + TILE * LDS_ROW;
  const uint32_t a0 = (uint32_t)(uint64_t)row0;  // raw LDS byte address for async VDST
  const uint32_t a1 = (uint32_t)(uint64_t)row1;

  // Issue 12 async 16B chunk copies: byte off of h_in row src[e] goes to the LDS address laddr + off
  auto stage = [&](int tile, uint32_t laddr) {
    int e = tile * TILE + lane;
    if (e >= n_edges) e = n_edges - 1;          // clamped lanes load valid memory
    uint32_t goff = (uint32_t)src[e] * ROW_BYTES;
#define STG(J)                                                              \
    asm volatile("global_load_async_to_lds_b128 %0, %1, %2 offset:" #J     \
                 :: "v"(laddr), "v"(goff), "s"(h_in) : "memory")
    STG(0);  STG(16);  STG(32);  STG(48);  STG(64);  STG(80);
    STG(96); STG(112); STG(128); STG(144); STG(160); STG(176);
#undef STG
  };

  auto consume = [&](int tile, AS3 char* rowp) {
    int e = tile * TILE + lane;
    if (e < n_edges) {
      float* outp = h_out + (uint64_t)dst[e] * D_FEAT;
      AS3 f4* r = (AS3 f4*)rowp;
#pragma unroll
      for (int j = 0; j < 12; ++j) {
        f4 v = r[j];
        if (RELU) {
          v.x = v.x > 0.0f ? v.x : 0.0f;
          v.y = v.y > 0.0f ? v.y : 0.0f;
          v.z = v.z > 0.0f ? v.z : 0.0f;
          v.w = v.w > 0.0f ? v.w : 0.0f;
        }
        atomicAdd(outp + j * 4 + 0, v.x);
        atomicAdd(outp + j * 4 + 1, v.y);
        atomicAdd(outp + j * 4 + 2, v.z);
        atomicAdd(outp + j * 4 + 3, v.w);
      }
    }
  };

  int t = wave;
  if (t >= n_tiles) return;
  stage(t, a0);                                  // prologue prefetch
  const uint32_t addrs[2] = {a0, a1};
  AS3 char* rows[2] = {row0, row1};
  int buf = 0;
  for (; t < n_tiles; t += n_waves) {
    int tn = t + n_waves;
    if (tn < n_tiles) {
      stage(tn, addrs[buf ^ 1]);                 // prefetch next tile
      // 24 async ops outstanding; async loads retire in order, so allowing
      // <=12 outstanding guarantees the current tile's 12 copies landed.
      asm volatile("s_wait_asynccnt 12" ::: "memory");
    } else {
      asm volatile("s_wait_asynccnt 0" ::: "memory");
    }
    consume(t, rows[buf]);
    buf ^= 1;
  }
}

// ------------------------------------------------------------------- launch
extern "C" void kernel_launch(void* const* d_in, const int* in_sizes, int n_in,
                              void* d_out, int out_size, void* d_ws, size_t ws_size,
                              hipStream_t stream) {
  (void)n_in; (void)out_size; (void)ws_size;
  const float* feat = (const float*)d_in[0];
  const int*   src  = (const int*)d_in[1];
  const int*   dst  = (const int*)d_in[2];
  float* out = (float*)d_out;
  float* ws  = (float*)d_ws;                     // needs n_nodes*48*4 = 19.2 MB

  const int n_elems = in_sizes[0];               // n_nodes * 48
  const int n_edges = in_sizes[1];
  const int n4 = n_elems / 4;

  const dim3 zgrid(2048), zblk(256);
  const dim3 egrid(2048), eblk(BLOCK);           // 8192 waves, ~6 tiles each

  // Layer 1: out = scatter(feat)          (relu deferred to next gather)
  gcn_zero<<<zgrid, zblk, 0, stream>>>((f4*)out, n4);
  gcn_edge_scatter<0><<<egrid, eblk, 0, stream>>>(feat, src, dst, out, n_edges);
  // Layer 2: ws = scatter(relu(out))
  gcn_zero<<<zgrid, zblk, 0, stream>>>((f4*)ws, n4);
  gcn_edge_scatter<1><<<egrid, eblk, 0, stream>>>(out, src, dst, ws, n_edges);
  // Layer 3: out = scatter(relu(ws))      (final layer: no activation)
  gcn_zero<<<zgrid, zblk, 0, stream>>>((f4*)out, n4);
  gcn_edge_scatter<1><<<egrid, eblk, 0, stream>>>(ws, src, dst, out, n_edges);
}